// GINLayer_53463752901319
// MI455X (gfx1250) — compile-verified
//
#include <hip/hip_runtime.h>

#define N_NODES 50000
#define N_EDGES 800000
#define DIM 128
#define ROWPAD 132   // padded LDS row stride (floats) to avoid bank conflicts

typedef float v2f __attribute__((ext_vector_type(2)));
typedef float v8f __attribute__((ext_vector_type(8)));

// ---------------------------------------------------------------------------
// Kernel 0: zero the accumulator (which is d_out itself).
// grid = 6250, block = 256  -> exactly 50000*128 floats as float4
// ---------------------------------------------------------------------------
__global__ __launch_bounds__(256) void gin_zero(float4* __restrict__ out) {
    int i = blockIdx.x * 256 + threadIdx.x;
    out[i] = float4{0.f, 0.f, 0.f, 0.f};
}

// ---------------------------------------------------------------------------
// Kernel 1: swizzle W1/W2 into WMMA B-fragment order so the GEMM inner loop
// does one coalesced 8-byte load per fragment per lane.
// Element (ks, jt, lane, c) = W[j*128 + k], j = jt*16 + (lane&15),
// k = ks*4 + (lane>>4)*2 + c.   16384 floats per weight matrix.
// ---------------------------------------------------------------------------
__global__ __launch_bounds__(256) void gin_swizzle(const float* __restrict__ W1,
                                                   const float* __restrict__ W2,
                                                   float* __restrict__ ws) {
    int idx = blockIdx.x * 256 + threadIdx.x;          // grid = 128 -> 32768
    int w    = idx >> 14;
    int i    = idx & 16383;
    int c    = i & 1;
    int lane = (i >> 1) & 31;
    int jt   = (i >> 6) & 7;
    int ks   = i >> 9;
    int mrow = lane & 15;
    int kofs = (lane >> 4) * 2;
    int j = jt * 16 + mrow;
    int k = ks * 4 + kofs + c;
    const float* W = w ? W2 : W1;
    ws[idx] = W[j * DIM + k];
}

// ---------------------------------------------------------------------------
// Kernel 2: edge aggregation. One wave per edge; each lane moves 4 floats.
// Gather is a coalesced float4 load (L2-resident table), scatter is four
// non-returning global_atomic_add_f32.
// ---------------------------------------------------------------------------
__global__ __launch_bounds__(256) void gin_edges(const float* __restrict__ x,
                                                 const int*  __restrict__ ei,
                                                 float* sum) {
    int e = blockIdx.x * 8 + (threadIdx.x >> 5);
    if (e >= N_EDGES) return;
    int lane = threadIdx.x & 31;
    int src = ei[e];
    int dst = ei[N_EDGES + e];
    const float4 v = *(const float4*)(x + (size_t)src * DIM + lane * 4);
    float* p = sum + (size_t)dst * DIM + lane * 4;
    __hip_atomic_fetch_add(p + 0, v.x, __ATOMIC_RELAXED, __HIP_MEMORY_SCOPE_AGENT);
    __hip_atomic_fetch_add(p + 1, v.y, __ATOMIC_RELAXED, __HIP_MEMORY_SCOPE_AGENT);
    __hip_atomic_fetch_add(p + 2, v.z, __ATOMIC_RELAXED, __HIP_MEMORY_SCOPE_AGENT);
    __hip_atomic_fetch_add(p + 3, v.w, __ATOMIC_RELAXED, __HIP_MEMORY_SCOPE_AGENT);
}

// ---------------------------------------------------------------------------
// Kernel 3: fused  z = (1+eta)*x + sum ; relu(z@W1^T+b1) ; relu(.@W2^T+b2) ;
//           LayerNorm ; + x.   One wave handles 16 rows (nodes).
// GEMMs use V_WMMA_F32_16X16X4_F32 (fp32 end to end).
// sumbuf aliases out (read-before-write, rows owned exclusively per wave).
// ---------------------------------------------------------------------------
__global__ __launch_bounds__(128) void gin_fused(const float* __restrict__ x,
                                                 const float* sumbuf,
                                                 const float* __restrict__ Ws1,
                                                 const float* __restrict__ Ws2,
                                                 const float* __restrict__ b1,
                                                 const float* __restrict__ b2,
                                                 const float* __restrict__ eta,
                                                 const float* __restrict__ gamma,
                                                 const float* __restrict__ beta,
                                                 float* out) {
    __shared__ float hbuf[4 * 16 * ROWPAD];   // per-wave 16x132 staging tile
    __shared__ float stats[4 * 16 * 2];       // per-row {mu, rsigma}

    const int lane = threadIdx.x & 31;
    const int wave = threadIdx.x >> 5;
    const int mrow = lane & 15;               // M index within 16-row tile
    const int half = lane >> 4;               // 0: lanes 0-15, 1: lanes 16-31
    const int kofs = half * 2;                // A/B fragment K offset
    float* hb = &hbuf[wave * 16 * ROWPAD];
    float* st = &stats[wave * 32];

    const int row0 = (blockIdx.x * 4 + wave) * 16;
    const int rowA = row0 + mrow;
    const int rA   = rowA < N_NODES ? rowA : (N_NODES - 1);  // clamp loads; EXEC stays full

    const float es = 1.0f + eta[0];

    // ---- Phase 1: build A-fragments of z in registers (full 16x128 tile) ----
    v2f afrag[32];
#pragma unroll
    for (int ks = 0; ks < 32; ++ks) {
        int col = ks * 4 + kofs;
        v2f xv = *(const v2f*)(x      + (size_t)rA * DIM + col);
        v2f sv = *(const v2f*)(sumbuf + (size_t)rA * DIM + col);
        afrag[ks] = es * xv + sv;
    }

    v8f acc[8];

    // ---- Phase 2: GEMM1 (z @ W1^T + b1), fp32 WMMA ----
#pragma unroll
    for (int jt = 0; jt < 8; ++jt) {
        float bv = b1[jt * 16 + mrow];
#pragma unroll
        for (int v = 0; v < 8; ++v) acc[jt][v] = bv;
    }
#pragma unroll
    for (int ks = 0; ks < 32; ++ks) {
        v2f a = afrag[ks];
#pragma unroll
        for (int jt = 0; jt < 8; ++jt) {
            v2f b = *(const v2f*)(Ws1 + (((ks * 8 + jt) * 32 + lane) << 1));
            acc[jt] = __builtin_amdgcn_wmma_f32_16x16x4_f32(
                false, a, false, b, (short)0, acc[jt], false, false);
        }
    }

    // ---- Phase 3: ReLU, C/D layout -> LDS (row-major, padded) ----
#pragma unroll
    for (int jt = 0; jt < 8; ++jt)
#pragma unroll
        for (int v = 0; v < 8; ++v) {
            float hv = acc[jt][v];
            hb[(v + half * 8) * ROWPAD + jt * 16 + mrow] = hv > 0.f ? hv : 0.f;
        }

    // ---- Phase 4: reload as A-fragments ----
#pragma unroll
    for (int ks = 0; ks < 32; ++ks)
        afrag[ks] = *(const v2f*)&hb[mrow * ROWPAD + ks * 4 + kofs];

    // ---- Phase 5: GEMM2 (h @ W2^T + b2), ReLU ----
#pragma unroll
    for (int jt = 0; jt < 8; ++jt) {
        float bv = b2[jt * 16 + mrow];
#pragma unroll
        for (int v = 0; v < 8; ++v) acc[jt][v] = bv;
    }
#pragma unroll
    for (int ks = 0; ks < 32; ++ks) {
        v2f a = afrag[ks];
#pragma unroll
        for (int jt = 0; jt < 8; ++jt) {
            v2f b = *(const v2f*)(Ws2 + (((ks * 8 + jt) * 32 + lane) << 1));
            acc[jt] = __builtin_amdgcn_wmma_f32_16x16x4_f32(
                false, a, false, b, (short)0, acc[jt], false, false);
        }
    }
#pragma unroll
    for (int jt = 0; jt < 8; ++jt)
#pragma unroll
        for (int v = 0; v < 8; ++v) {
            float hv = acc[jt][v];
            hb[(v + half * 8) * ROWPAD + jt * 16 + mrow] = hv > 0.f ? hv : 0.f;
        }

    // ---- Phase 6: LayerNorm stats. Lane pair (mrow, mrow+16) owns one row. ----
    {
        float s1 = 0.f, s2 = 0.f;
        const float* rowp = &hb[mrow * ROWPAD + half * 64];
#pragma unroll
        for (int c = 0; c < 64; ++c) {
            float v = rowp[c];
            s1 += v;
            s2 += v * v;
        }
        s1 += __shfl_xor(s1, 16, 32);
        s2 += __shfl_xor(s2, 16, 32);
        float mu  = s1 * (1.0f / 128.0f);
        float var = s2 * (1.0f / 128.0f) - mu * mu;
        float rs  = rsqrtf(var + 1e-5f);
        if (half == 0) {
            st[mrow * 2 + 0] = mu;
            st[mrow * 2 + 1] = rs;
        }
    }

    // ---- Phase 7: normalize + affine + skip, coalesced float4 stores ----
    const int col = lane * 4;
    const float4 g  = *(const float4*)(gamma + col);
    const float4 be = *(const float4*)(beta + col);
    for (int r = 0; r < 16; ++r) {
        int grow = row0 + r;
        if (grow >= N_NODES) break;
        float mu = st[r * 2 + 0];
        float rs = st[r * 2 + 1];
        const float* hp = &hb[r * ROWPAD + col];
        const float4 xv = *(const float4*)(x + (size_t)grow * DIM + col);
        float4 o;
        o.x = (hp[0] - mu) * rs * g.x + be.x + xv.x;
        o.y = (hp[1] - mu) * rs * g.y + be.y + xv.y;
        o.z = (hp[2] - mu) * rs * g.z + be.z + xv.z;
        o.w = (hp[3] - mu) * rs * g.w + be.w + xv.w;
        *(float4*)(out + (size_t)grow * DIM + col) = o;
    }
}

// ---------------------------------------------------------------------------
extern "C" void kernel_launch(void* const* d_in, const int* in_sizes, int n_in,
                              void* d_out, int out_size, void* d_ws, size_t ws_size,
                              hipStream_t stream) {
    const float* x     = (const float*)d_in[0];
    const int*   ei    = (const int*)d_in[1];
    const float* W1    = (const float*)d_in[2];
    const float* b1    = (const float*)d_in[3];
    const float* W2    = (const float*)d_in[4];
    const float* b2    = (const float*)d_in[5];
    const float* eta   = (const float*)d_in[6];
    const float* gamma = (const float*)d_in[7];
    const float* beta  = (const float*)d_in[8];
    float* out = (float*)d_out;
    float* wsf = (float*)d_ws;          // needs 2*16384 floats = 128 KB
    float* Ws1 = wsf;
    float* Ws2 = wsf + 16384;

    // 1) zero accumulator (d_out doubles as segment-sum buffer)
    gin_zero<<<(N_NODES * DIM / 4) / 256, 256, 0, stream>>>((float4*)out);
    // 2) swizzle weights into B-fragment order (independent of 1)
    gin_swizzle<<<(2 * 16384) / 256, 256, 0, stream>>>(W1, W2, wsf);
    // 3) edge gather + atomic segment-sum into d_out
    gin_edges<<<N_EDGES / 8, 256, 0, stream>>>(x, ei, out);
    // 4) fused z -> MLP (fp32 WMMA) -> LayerNorm -> skip, in-place over d_out
    int nwaves  = (N_NODES + 15) / 16;          // 3125
    int nblocks = (nwaves + 3) / 4;             // 782
    gin_fused<<<nblocks, 128, 0, stream>>>(x, out, Ws1, Ws2, b1, b2, eta,
                                           gamma, beta, out);
}